// QuantumGate_65481071399866
// MI455X (gfx1250) — compile-verified
//
#include <hip/hip_runtime.h>
#include <math.h>

typedef __attribute__((ext_vector_type(2))) float v2f;
typedef __attribute__((ext_vector_type(8))) float v8f;

#define N_Q   14
#define BATCH 256

// One wave32 per block; block b owns batch rows [16b, 16b+16).
// Fused: angles = x@W^T (WMMA) -> cos -> closed-form prefix products -> out = z@W^T (WMMA).
// x and W are staged through zero-padded 16x16 LDS tiles so every WMMA fragment
// read is an unconditional 8B-aligned ds_load_b64 (no divergent bounds checks).
__global__ __launch_bounds__(32) void qgate_fused_kernel(const float* __restrict__ x,
                                                         const float* __restrict__ W,
                                                         float* __restrict__ out) {
    __shared__ __align__(16) float xt[16][16];  // x tile / angles / z tile (padded)
    __shared__ __align__(16) float wt[16][16];  // W padded: wt[n][k] = W[n][k]

    const int lane  = threadIdx.x;   // 0..31 (wave32)
    const int half  = lane >> 4;     // which K-pair group this lane holds
    const int ln    = lane & 15;     // M for A-frags, N for B/C/D-frags
    const int mbase = blockIdx.x * 16;

    // ---- Zero both tiles (covers the pad rows/cols), then cooperative fill.
    float* xf = &xt[0][0];
    float* wf = &wt[0][0];
#pragma unroll
    for (int i = 0; i < 8; ++i) {
        xf[lane + 32 * i] = 0.0f;
        wf[lane + 32 * i] = 0.0f;
    }
    __syncthreads();

    // x tile: 16 rows x 14 cols = 224 floats, coalesced flat loads.
#pragma unroll
    for (int i = 0; i < 7; ++i) {
        const int idx = lane + 32 * i;            // 0..223
        xt[idx / N_Q][idx % N_Q] = x[mbase * N_Q + idx];
    }
    // W: 196 floats = 6 full strips + 4 leftover elements.
#pragma unroll
    for (int i = 0; i < 6; ++i) {
        const int idx = lane + 32 * i;            // 0..191
        wt[idx / N_Q][idx % N_Q] = W[idx];
    }
    if (lane < 4) {
        const int idx = 192 + lane;               // 192..195
        wt[idx / N_Q][idx % N_Q] = W[idx];
    }
    __syncthreads();

    // ---- B fragments: B = W^T.  Lane holds {B[k][ln], B[k+1][ln]} = {wt[ln][k], wt[ln][k+1]},
    // k = 4t + 2*half  ->  consecutive LDS floats, 8B aligned  ->  ds_load_b64.
    v2f bfrag[4];
#pragma unroll
    for (int t = 0; t < 4; ++t) {
        const int k = t * 4 + half * 2;
        bfrag[t] = *(const v2f*)&wt[ln][k];
    }

    // ---- GEMM 1: angles = x @ W^T  (four K=4 WMMA steps, f32 end to end)
    v8f acc = {};
#pragma unroll
    for (int t = 0; t < 4; ++t) {
        const int k = t * 4 + half * 2;
        v2f afrag = *(const v2f*)&xt[ln][k];
        acc = __builtin_amdgcn_wmma_f32_16x16x4_f32(
                  /*neg_a=*/false, afrag, /*neg_b=*/false, bfrag[t],
                  /*c_mod=*/(short)0, acc, /*reuse_a=*/false, /*reuse_b=*/false);
    }

    // acc[v] = angles[M = v + 8*half][N = ln].  cos elementwise (all 32 lanes busy),
    // park back into the x tile so each lane can own a full row for the prefix product.
#pragma unroll
    for (int v = 0; v < 8; ++v)
        xt[v + 8 * half][ln] = __cosf(acc[v]);
    __syncthreads();

    // ---- Closed-form circuit: z_j = prod_{i<=j} cos(theta_i) (j>=1),
    //      z_0 = prod_{i=1..13} cos(theta_i).  Lanes 0-15: one batch row each.
    if (lane < 16) {
        float c[N_Q];
#pragma unroll
        for (int j = 0; j < N_Q; ++j) c[j] = xt[lane][j];

        float z[16];
        float p = c[0];
#pragma unroll
        for (int j = 1; j < N_Q; ++j) { p *= c[j]; z[j] = p; }
        float q = 1.0f;
#pragma unroll
        for (int j = 1; j < N_Q; ++j) q *= c[j];
        z[0]  = q;
        z[14] = 0.0f;   // re-zero K padding (cos turned it into 1.0)
        z[15] = 0.0f;
#pragma unroll
        for (int j = 0; j < 16; ++j) xt[lane][j] = z[j];
    }
    __syncthreads();

    // ---- GEMM 2: out = z @ W^T  (reuse the W^T B-fragments already in VGPRs)
    v8f acc2 = {};
#pragma unroll
    for (int t = 0; t < 4; ++t) {
        const int k = t * 4 + half * 2;
        v2f afrag = *(const v2f*)&xt[ln][k];
        acc2 = __builtin_amdgcn_wmma_f32_16x16x4_f32(
                   false, afrag, false, bfrag[t],
                   (short)0, acc2, false, false);
    }

    // D layout: lane holds out[M = v + 8*half][N = ln]; only N < 14 is real.
    if (ln < N_Q) {
#pragma unroll
        for (int v = 0; v < 8; ++v)
            out[(mbase + v + 8 * half) * N_Q + ln] = acc2[v];
    }
}

extern "C" void kernel_launch(void* const* d_in, const int* in_sizes, int n_in,
                              void* d_out, int out_size, void* d_ws, size_t ws_size,
                              hipStream_t stream) {
    const float* x = (const float*)d_in[0];   // (256, 14) f32
    const float* W = (const float*)d_in[1];   // (14, 14)  f32
    float* out = (float*)d_out;               // (256, 14) f32
    (void)in_sizes; (void)n_in; (void)out_size; (void)d_ws; (void)ws_size;

    qgate_fused_kernel<<<dim3(BATCH / 16), dim3(32), 0, stream>>>(x, W, out);
}